// WeightedDualEdgePredictor_55980603736434
// MI455X (gfx1250) — compile-verified
//
#include <hip/hip_runtime.h>
#include <hip/hip_bf16.h>
#include <math.h>

#define NN 1024
#define NE 16384
#define FIN 128
#define HH 64
#define ALPHA_C 0.7f

typedef float v2f __attribute__((ext_vector_type(2)));
typedef float v8f __attribute__((ext_vector_type(8)));

// ---------------------------------------------------------------------------
// fp32 WMMA GEMM: C[M,Nc] = A[M,K] @ W[K,Nc]   (row-major, Nc multiple of 16,
// K multiple of 4, M multiple of 16). One wave computes one 16x16 C tile via
// V_WMMA_F32_16X16X4_F32, accumulating over K in steps of 4.
//
// VGPR layouts per CDNA5 ISA 7.12.2:
//   A 16x4 f32 : lanes 0-15 -> M=lane, {K=0,K=1}; lanes 16-31 -> {K=2,K=3}
//   B 4x16 f32 : lanes 0-15 -> N=lane, {K=0,K=1}; lanes 16-31 -> {K=2,K=3}
//   C 16x16 f32: VGPR v, lanes 0-15 -> M=v, lanes 16-31 -> M=v+8; N=lane%16
// ---------------------------------------------------------------------------
__global__ __launch_bounds__(256)
void gemm_wmma_f32(const float* __restrict__ A, const float* __restrict__ W,
                   float* __restrict__ C, int M, int K, int Nc) {
  const int lane = threadIdx.x;          // 0..31
  const int wave = threadIdx.y;          // 0..7
  const int tiles_n = Nc >> 4;
  const int tile = blockIdx.x * blockDim.y + wave;
  const int tm = (tile / tiles_n) << 4;
  const int tn = (tile % tiles_n) << 4;
  if (tm >= M) return;                   // uniform per wave -> EXEC stays all-1s

  const int half = lane >> 4;            // 0: K pair {0,1}; 1: K pair {2,3}
  const int l16  = lane & 15;

  v8f acc = {};
  const float* __restrict__ Arow = A + (size_t)(tm + l16) * K;
  #pragma unroll 4
  for (int k = 0; k < K; k += 4) {
    const int kk = k + 2 * half;
    v2f a, b;
    a.x = Arow[kk];
    a.y = Arow[kk + 1];
    b.x = W[(size_t)kk       * Nc + tn + l16];
    b.y = W[(size_t)(kk + 1) * Nc + tn + l16];
    acc = __builtin_amdgcn_wmma_f32_16x16x4_f32(
        /*neg_a=*/false, a, /*neg_b=*/false, b,
        /*c_mod=*/(short)0, acc, /*reuse_a=*/false, /*reuse_b=*/false);
  }
  #pragma unroll
  for (int v = 0; v < 8; ++v) {
    const int row = tm + v + 8 * half;
    C[(size_t)row * Nc + tn + l16] = acc[v];
  }
}

// ---------------------------------------------------------------------------
// Utility / graph kernels
// ---------------------------------------------------------------------------
__global__ void fill_kernel(float* p, int n, float v) {
  int i = blockIdx.x * blockDim.x + threadIdx.x;
  if (i < n) p[i] = v;
}

__global__ void deg_kernel(const int* __restrict__ dst, float* __restrict__ deg) {
  int e = blockIdx.x * blockDim.x + threadIdx.x;
  if (e < NE) atomicAdd(&deg[dst[e]], 1.0f);
}

__global__ void dinv_kernel(const float* __restrict__ deg, float* __restrict__ dinv) {
  int i = blockIdx.x * blockDim.x + threadIdx.x;
  if (i < NN) {
    float d = deg[i] + 1.0f;               // +1 self loop
    dinv[i] = (d > 0.0f) ? rsqrtf(d) : 0.0f;
  }
}

// one block per (edge or self-loop), 64 threads = one feature each
__global__ __launch_bounds__(64)
void gcn_scatter(const int* __restrict__ src, const int* __restrict__ dst,
                 const float* __restrict__ dinv, const float* __restrict__ h,
                 float* __restrict__ acc) {
  int idx = blockIdx.x;                    // 0 .. NE+NN-1
  int f = threadIdx.x;
  int s, d;
  if (idx < NE) { s = src[idx]; d = dst[idx]; }
  else          { s = d = idx - NE; }
  float norm = dinv[s] * dinv[d];
  atomicAdd(&acc[(size_t)d * HH + f], norm * h[(size_t)s * HH + f]);
}

__global__ void bias_relu_kernel(const float* __restrict__ acc,
                                 const float* __restrict__ b,
                                 float* __restrict__ out, int n) {
  int i = blockIdx.x * blockDim.x + threadIdx.x;
  if (i < n) out[i] = fmaxf(acc[i] + b[i & (HH - 1)], 0.0f);
}

__global__ void combine_kernel(const float* __restrict__ xa,
                               const float* __restrict__ xd,
                               float* __restrict__ xc, int n) {
  int i = blockIdx.x * blockDim.x + threadIdx.x;
  if (i < n) xc[i] = ALPHA_C * xa[i] + (1.0f - ALPHA_C) * xd[i];
}

// grid-stride sum reduction -> atomicAdd into *out (caller zero-inits *out)
__global__ void reduce_sum_kernel(const float* __restrict__ x, int n, float* out) {
  __shared__ float sm[256];
  float s = 0.0f;
  for (int i = blockIdx.x * blockDim.x + threadIdx.x; i < n;
       i += gridDim.x * blockDim.x) s += x[i];
  sm[threadIdx.x] = s;
  __syncthreads();
  for (int off = 128; off > 0; off >>= 1) {
    if ((int)threadIdx.x < off) sm[threadIdx.x] += sm[threadIdx.x + off];
    __syncthreads();
  }
  if (threadIdx.x == 0) atomicAdd(out, sm[0]);
}

// dot of two length-64 vectors, one wave32
__global__ __launch_bounds__(32)
void dot64_kernel(const float* __restrict__ a, const float* __restrict__ b,
                  float* __restrict__ out) {
  int lane = threadIdx.x;
  float s = a[lane] * b[lane] + a[lane + 32] * b[lane + 32];
  #pragma unroll
  for (int off = 16; off > 0; off >>= 1) s += __shfl_down(s, off, 32);
  if (lane == 0) *out = s;
}

// per-node attention logits: al_s = h@a_src, al_d = h@a_dst (one wave / node)
__global__ __launch_bounds__(32)
void gat_node_alpha(const float* __restrict__ h, const float* __restrict__ a_src,
                    const float* __restrict__ a_dst,
                    float* __restrict__ al_s, float* __restrict__ al_d) {
  int i = blockIdx.x;
  int lane = threadIdx.x;
  float h0 = h[(size_t)i * HH + lane];
  float h1 = h[(size_t)i * HH + lane + 32];
  float vs = h0 * a_src[lane] + h1 * a_src[lane + 32];
  float vd = h0 * a_dst[lane] + h1 * a_dst[lane + 32];
  #pragma unroll
  for (int off = 16; off > 0; off >>= 1) {
    vs += __shfl_down(vs, off, 32);
    vd += __shfl_down(vd, off, 32);
  }
  if (lane == 0) { al_s[i] = vs; al_d[i] = vd; }
}

__device__ __forceinline__ void atomicMaxFloat(float* addr, float val) {
  // works for mixed signs with init = -inf (0xFF800000)
  if (val >= 0.0f) atomicMax((int*)addr, __float_as_int(val));
  else             atomicMin((unsigned int*)addr, __float_as_uint(val));
}

__global__ void gat_edge_score(const int* __restrict__ src, const int* __restrict__ dst,
                               const float* __restrict__ ew,
                               const float* __restrict__ al_s, const float* __restrict__ al_d,
                               const float* __restrict__ c_scal,
                               const float* __restrict__ ew_sum,
                               float* __restrict__ e_arr, float* __restrict__ emax) {
  int idx = blockIdx.x * blockDim.x + threadIdx.x;
  if (idx >= NE + NN) return;
  int s, d; float w;
  if (idx < NE) { s = src[idx]; d = dst[idx]; w = ew[idx]; }
  else          { s = d = idx - NE; w = (*ew_sum) * (1.0f / (float)NE); }
  float e = al_s[s] + al_d[d] + w * (*c_scal);
  e = (e > 0.0f) ? e : 0.2f * e;           // leaky_relu(0.2)
  e_arr[idx] = e;
  atomicMaxFloat(&emax[d], e);
}

__global__ void gat_exp_kernel(const int* __restrict__ dst,
                               const float* __restrict__ e_arr,
                               const float* __restrict__ emax,
                               float* __restrict__ ex_arr, float* __restrict__ denom) {
  int idx = blockIdx.x * blockDim.x + threadIdx.x;
  if (idx >= NE + NN) return;
  int d = (idx < NE) ? dst[idx] : (idx - NE);
  float ex = expf(e_arr[idx] - emax[d]);
  ex_arr[idx] = ex;
  atomicAdd(&denom[d], ex);
}

__global__ __launch_bounds__(64)
void gat_scatter(const int* __restrict__ src, const int* __restrict__ dst,
                 const float* __restrict__ ex_arr, const float* __restrict__ denom,
                 const float* __restrict__ h, float* __restrict__ acc) {
  int idx = blockIdx.x;
  int f = threadIdx.x;
  int s, d;
  if (idx < NE) { s = src[idx]; d = dst[idx]; }
  else          { s = d = idx - NE; }
  float alpha = ex_arr[idx] / denom[d];
  atomicAdd(&acc[(size_t)d * HH + f], alpha * h[(size_t)s * HH + f]);
}

// ---------------------------------------------------------------------------
// All-pairs predictor: out[i,j] = sigmoid( sum_h relu(h1[i,h]+h2[j,h]+bp1[h])
//                                          * wp2[h] + bp2 )
// 32x32 output tile per block; tiles staged in LDS (+1 pad, conflict-free);
// each thread computes 4 rows x 1 col, reusing each h2 LDS read 4x.
// ---------------------------------------------------------------------------
__global__ __launch_bounds__(256)
void predictor_kernel(const float* __restrict__ h1, const float* __restrict__ h2,
                      const float* __restrict__ bp1, const float* __restrict__ wp2,
                      const float* __restrict__ bp2, float* __restrict__ out) {
  __shared__ float s1[32][HH + 1];
  __shared__ float s2[32][HH + 1];
  __shared__ float sw[HH];
  const int tid = threadIdx.x;
  const int i0 = blockIdx.y << 5;
  const int j0 = blockIdx.x << 5;
  for (int t = tid; t < 32 * HH; t += 256) {
    int r = t >> 6, f = t & (HH - 1);
    s1[r][f] = h1[(size_t)(i0 + r) * HH + f] + bp1[f];   // fold bias into h1 tile
    s2[r][f] = h2[(size_t)(j0 + r) * HH + f];
  }
  if (tid < HH) sw[tid] = wp2[tid];
  __syncthreads();

  const float b2 = bp2[0];
  const int c  = tid & 31;    // column in tile (lane -> broadcast-friendly s1)
  const int r0 = tid >> 5;    // 0..7
  float a0 = 0.f, a1 = 0.f, a2 = 0.f, a3 = 0.f;
  #pragma unroll 8
  for (int f = 0; f < HH; ++f) {
    const float v2 = s2[c][f];
    const float w  = sw[f];
    a0 += fmaxf(s1[r0     ][f] + v2, 0.0f) * w;
    a1 += fmaxf(s1[r0 +  8][f] + v2, 0.0f) * w;
    a2 += fmaxf(s1[r0 + 16][f] + v2, 0.0f) * w;
    a3 += fmaxf(s1[r0 + 24][f] + v2, 0.0f) * w;
  }
  out[(size_t)(i0 + r0     ) * NN + j0 + c] = 1.0f / (1.0f + expf(-(a0 + b2)));
  out[(size_t)(i0 + r0 +  8) * NN + j0 + c] = 1.0f / (1.0f + expf(-(a1 + b2)));
  out[(size_t)(i0 + r0 + 16) * NN + j0 + c] = 1.0f / (1.0f + expf(-(a2 + b2)));
  out[(size_t)(i0 + r0 + 24) * NN + j0 + c] = 1.0f / (1.0f + expf(-(a3 + b2)));
}

// ---------------------------------------------------------------------------
extern "C" void kernel_launch(void* const* d_in, const int* in_sizes, int n_in,
                              void* d_out, int out_size, void* d_ws, size_t ws_size,
                              hipStream_t stream) {
  const float* x        = (const float*)d_in[0];
  const int*   eia      = (const int*)d_in[1];   // [2,E] above
  const int*   eid      = (const int*)d_in[2];   // [2,E] distance
  const float* ew       = (const float*)d_in[3];
  const float* W_gcn1   = (const float*)d_in[4];
  const float* b_gcn1   = (const float*)d_in[5];
  const float* W_gat1   = (const float*)d_in[6];
  const float* att_src1 = (const float*)d_in[7];
  const float* att_dst1 = (const float*)d_in[8];
  const float* w_edge1  = (const float*)d_in[9];
  const float* att_edge1= (const float*)d_in[10];
  const float* b_gat1   = (const float*)d_in[11];
  const float* W_gcn2   = (const float*)d_in[12];
  const float* b_gcn2   = (const float*)d_in[13];
  const float* W_gat2   = (const float*)d_in[14];
  const float* att_src2 = (const float*)d_in[15];
  const float* att_dst2 = (const float*)d_in[16];
  const float* w_edge2  = (const float*)d_in[17];
  const float* att_edge2= (const float*)d_in[18];
  const float* b_gat2   = (const float*)d_in[19];
  const float* W_gcn3   = (const float*)d_in[20];
  const float* b_gcn3   = (const float*)d_in[21];
  const float* Wp1      = (const float*)d_in[22];
  const float* bp1      = (const float*)d_in[23];
  const float* Wp2      = (const float*)d_in[24];
  const float* bp2      = (const float*)d_in[25];
  float* out = (float*)d_out;

  const int* sa = eia;            const int* da = eia + NE;
  const int* sd = eid;            const int* dd = eid + NE;

  // workspace layout (floats)
  const size_t NH = (size_t)NN * HH;
  float* ws   = (float*)d_ws;
  float* hg   = ws + 0 * NH;      // gemm out (GCN path)
  float* ht   = ws + 1 * NH;      // gemm out (GAT path)
  float* accg = ws + 2 * NH;      // GCN scatter accumulator
  float* acct = ws + 3 * NH;      // GAT scatter accumulator
  float* xa   = ws + 4 * NH;      // GCN activations
  float* xd   = ws + 5 * NH;      // GAT activations
  float* xc   = ws + 6 * NH;
  float* hf   = ws + 7 * NH;      // final node embedding
  float* h1   = ws + 8 * NH;
  float* h2   = ws + 9 * NH;
  float* deg  = ws + 10 * NH;
  float* dinv = deg + NN;
  float* als  = dinv + NN;
  float* ald  = als + NN;
  float* emax = ald + NN;
  float* denom= emax + NN;
  float* scal = denom + NN;               // [0]=sum(ew), [1]=c
  float* e_arr  = scal + 8;
  float* ex_arr = e_arr + (NE + NN);

  const dim3 gemmBlk(32, 8);
  auto gemm = [&](const float* A, const float* W, float* C, int M, int K, int Nc) {
    int tiles = (M / 16) * (Nc / 16);
    gemm_wmma_f32<<<dim3((tiles + 7) / 8), gemmBlk, 0, stream>>>(A, W, C, M, K, Nc);
  };
  auto fill = [&](float* p, int n, float v) {
    fill_kernel<<<(n + 255) / 256, 256, 0, stream>>>(p, n, v);
  };

  const int EN = NE + NN;
  const int enBlocks = (EN + 255) / 256;

  // --- shared precomputation ------------------------------------------------
  fill(deg, NN, 0.0f);
  deg_kernel<<<(NE + 255) / 256, 256, 0, stream>>>(da, deg);
  dinv_kernel<<<(NN + 255) / 256, 256, 0, stream>>>(deg, dinv);
  fill(scal, 8, 0.0f);
  reduce_sum_kernel<<<64, 256, 0, stream>>>(ew, NE, &scal[0]);

  // --- layer 1 GCN (above edges): x[1024x128] @ W_gcn1 ----------------------
  gemm(x, W_gcn1, hg, NN, FIN, HH);
  fill(accg, (int)NH, 0.0f);
  gcn_scatter<<<EN, 64, 0, stream>>>(sa, da, dinv, hg, accg);
  bias_relu_kernel<<<(int)(NH + 255) / 256, 256, 0, stream>>>(accg, b_gcn1, xa, (int)NH);

  // --- layer 1 GAT (distance edges) ----------------------------------------
  gemm(x, W_gat1, ht, NN, FIN, HH);
  dot64_kernel<<<1, 32, 0, stream>>>(w_edge1, att_edge1, &scal[1]);
  gat_node_alpha<<<NN, 32, 0, stream>>>(ht, att_src1, att_dst1, als, ald);
  fill(emax, NN, -INFINITY);
  gat_edge_score<<<enBlocks, 256, 0, stream>>>(sd, dd, ew, als, ald, &scal[1], &scal[0], e_arr, emax);
  fill(denom, NN, 0.0f);
  gat_exp_kernel<<<enBlocks, 256, 0, stream>>>(dd, e_arr, emax, ex_arr, denom);
  fill(acct, (int)NH, 0.0f);
  gat_scatter<<<EN, 64, 0, stream>>>(sd, dd, ex_arr, denom, ht, acct);
  bias_relu_kernel<<<(int)(NH + 255) / 256, 256, 0, stream>>>(acct, b_gat1, xd, (int)NH);

  // --- layer 2 GCN ----------------------------------------------------------
  gemm(xa, W_gcn2, hg, NN, HH, HH);
  fill(accg, (int)NH, 0.0f);
  gcn_scatter<<<EN, 64, 0, stream>>>(sa, da, dinv, hg, accg);
  bias_relu_kernel<<<(int)(NH + 255) / 256, 256, 0, stream>>>(accg, b_gcn2, xa, (int)NH);

  // --- layer 2 GAT ----------------------------------------------------------
  gemm(xd, W_gat2, ht, NN, HH, HH);
  dot64_kernel<<<1, 32, 0, stream>>>(w_edge2, att_edge2, &scal[1]);
  gat_node_alpha<<<NN, 32, 0, stream>>>(ht, att_src2, att_dst2, als, ald);
  fill(emax, NN, -INFINITY);
  gat_edge_score<<<enBlocks, 256, 0, stream>>>(sd, dd, ew, als, ald, &scal[1], &scal[0], e_arr, emax);
  fill(denom, NN, 0.0f);
  gat_exp_kernel<<<enBlocks, 256, 0, stream>>>(dd, e_arr, emax, ex_arr, denom);
  fill(acct, (int)NH, 0.0f);
  gat_scatter<<<EN, 64, 0, stream>>>(sd, dd, ex_arr, denom, ht, acct);
  bias_relu_kernel<<<(int)(NH + 255) / 256, 256, 0, stream>>>(acct, b_gat2, xd, (int)NH);

  // --- combine + layer 3 GCN ------------------------------------------------
  combine_kernel<<<(int)(NH + 255) / 256, 256, 0, stream>>>(xa, xd, xc, (int)NH);
  gemm(xc, W_gcn3, hg, NN, HH, HH);
  fill(accg, (int)NH, 0.0f);
  gcn_scatter<<<EN, 64, 0, stream>>>(sa, da, dinv, hg, accg);
  bias_relu_kernel<<<(int)(NH + 255) / 256, 256, 0, stream>>>(accg, b_gcn3, hf, (int)NH);

  // --- predictor ------------------------------------------------------------
  gemm(hf, Wp1,           h1, NN, HH, HH);   // h @ Wp1[:H]
  gemm(hf, Wp1 + HH * HH, h2, NN, HH, HH);   // h @ Wp1[H:]
  predictor_kernel<<<dim3(NN / 32, NN / 32), 256, 0, stream>>>(h1, h2, bp1, Wp2, bp2, out);
}